// Deemphasis_83167746720080
// MI455X (gfx1250) — compile-verified
//
#include <hip/hip_runtime.h>
#include <hip/hip_bf16.h>
#include <math.h>

// De-emphasis y[n] = x[n] + 0.97*y[n-1] over 64 rows x 480000 fp32 samples.
//
// Memory-bound: ~246 MB minimum traffic -> ~10.5us at 23.3 TB/s; halo reads
// add ~16% -> ~11.4us projected. Each wave owns a 6400-sample chunk and warms
// its carry over a 1024-sample halo (c^1024 ~ 3e-14, far below fp32 ulp of y),
// so all 4800 waves run fully parallel. Inside a 256-sample tile the scan is
// Y = L @ X with L[i][j] = c^(i-j) lower-triangular, computed by four chained
// V_WMMA_F32_16X16X4_F32 (the fp32 WMMA shape on CDNA5). Cross-segment
// carries: 16-wide ds_bpermute scan with ratio c^16. Cross-tile carry: scalar
// chain in the wave.
//
// Round-3 fix: __builtin_nontemporal_store requires a clang ext_vector type,
// not HIP's float4 struct -> store through a v4f (ext_vector_type(4) float).

typedef __attribute__((ext_vector_type(2))) float v2f;
typedef __attribute__((ext_vector_type(4))) float v4f;
typedef __attribute__((ext_vector_type(8))) float v8f;

#define COEFF 0.97f
#define N_SAMP 480000
#define ROWS 64
#define TILE 256
#define TILES_PER_ROW (N_SAMP / TILE)                   // 1875
#define WAVES_PER_ROW 75
#define TILES_PER_WAVE (TILES_PER_ROW / WAVES_PER_ROW)  // 25 tiles = 6400 samples
#define HALO_TILES 4                                    // 1024-sample warmup
#define WAVES_PER_BLOCK 8
#define BLOCK_THREADS (WAVES_PER_BLOCK * 32)
#define TOTAL_WAVES (ROWS * WAVES_PER_ROW)              // 4800
#define GRID_BLOCKS (TOTAL_WAVES / WAVES_PER_BLOCK)     // 600

__global__ __launch_bounds__(BLOCK_THREADS)
void deemph_scan_wmma(const float* __restrict__ x, float* __restrict__ y) {
    const int lane        = threadIdx.x & 31;
    const int wave_in_blk = threadIdx.x >> 5;
    const int gw          = blockIdx.x * WAVES_PER_BLOCK + wave_in_blk;
    const int row         = gw / WAVES_PER_ROW;
    const int wchunk      = gw % WAVES_PER_ROW;

    const int col  = lane & 15;   // segment (column) this lane serves in B/C/D
    const int half = lane >> 4;   // lane-half selects interleaved K / M rows

    // ---- A operand: L[i][j] = c^(i-j) for i>=j, split into 4 (16x4) k-blocks.
    // A 16x4 layout: VGPR0 lanes0-15 -> K=k0, lanes16-31 -> K=k0+2;
    //                VGPR1 -> K=k0+1 / k0+3.  (k0 = 4r)
    v2f aop[4];
#pragma unroll
    for (int r = 0; r < 4; ++r) {
        const int k0 = 4 * r + 2 * half;
        aop[r][0] = (col >= k0)     ? powf(COEFF, (float)(col - k0))     : 0.0f;
        aop[r][1] = (col >= k0 + 1) ? powf(COEFF, (float)(col - k0 - 1)) : 0.0f;
    }
    const float c16  = powf(COEFF, 16.0f);           // segment-to-segment ratio
    const float c256 = powf(COEFF, 256.0f);          // tile-to-tile ratio
    const float c16p = powf(c16, (float)col);        // c^(16*col)
    float cp[8];                                     // c^(i+1), i = row in C/D
#pragma unroll
    for (int t = 0; t < 8; ++t) cp[t] = powf(COEFF, (float)(t + 8 * half + 1));

    const float* __restrict__ xrow = x + (size_t)row * N_SAMP;
    float*       __restrict__ yrow = y + (size_t)row * N_SAMP;

    const int first_tile = wchunk * TILES_PER_WAVE;
    const int last_tile  = first_tile + TILES_PER_WAVE;  // exclusive
    int start_tile = first_tile - HALO_TILES;
    if (start_tile < 0) start_tile = 0;

    // Per-lane load offset inside a tile (one float2 per k-block).
    const int lofs = 16 * col + 2 * half;

    // ---- Prologue: load the first tile's B operands.
    v2f bcur[4];
#pragma unroll
    for (int r = 0; r < 4; ++r) {
        const float2 q = *reinterpret_cast<const float2*>(
            xrow + start_tile * TILE + lofs + 4 * r);
        bcur[r][0] = q.x;
        bcur[r][1] = q.y;
    }

    float carry = 0.0f;  // y-value just before current tile (uniform per wave)
    for (int tile = start_tile; tile < last_tile; ++tile) {
        // ---- Prefetch tile t+1 (clamped; redundant re-load on last iter keeps
        // control flow straight-line and in-bounds). Issued before the WMMAs so
        // the load clause is in flight during WMMA + bpermute-scan latency.
        const int ntile = (tile + 1 < last_tile) ? (tile + 1) : tile;
        v2f bnxt[4];
#pragma unroll
        for (int r = 0; r < 4; ++r) {
            const float2 q = *reinterpret_cast<const float2*>(
                xrow + ntile * TILE + lofs + 4 * r);
            bnxt[r][0] = q.x;
            bnxt[r][1] = q.y;
        }

        // ---- Y = L @ X : 4 chained fp32 WMMAs (accumulate through C).
        v8f d = {};
#pragma unroll
        for (int r = 0; r < 4; ++r) {
            d = __builtin_amdgcn_wmma_f32_16x16x4_f32(
                /*neg_a=*/false, aop[r], /*neg_b=*/false, bcur[r],
                /*c_mod=*/(short)0, d, /*reuse_a=*/false, /*reuse_b=*/false);
        }

        // ---- Cross-segment carries inside the tile.
        // t_j = Y[15][j] lives in lanes 16..31 of d[7]; broadcast to all lanes.
        float tj = __shfl(d[7], 16 + col);
        // Inclusive Hillis-Steele scan over j with ratio c16 (halves mirrored,
        // so plain lane-offset shuffles never cross halves when gated).
        float s    = tj;
        float wmul = c16;
#pragma unroll
        for (int off = 1; off <= 8; off <<= 1) {
            float o = __shfl(s, (lane - off) & 31);
            if (col >= off) s = __builtin_fmaf(o, wmul, s);
            wmul *= wmul;
        }
        // Carry into segment j: P_j = (j? S_{j-1} : 0) + c^(16j) * carry.
        float sprev = __shfl(s, (lane - 1) & 31);
        float P     = ((col == 0) ? 0.0f : sprev) + c16p * carry;
        // Carry out of this tile: S_15 + c^256 * carry.
        float s15 = __shfl(s, 15);
        carry     = __builtin_fmaf(c256, carry, s15);

        // ---- Apply carries and store nontemporally (skip halo tiles).
        if (tile >= first_tile) {
            float o[8];
#pragma unroll
            for (int t = 0; t < 8; ++t) o[t] = __builtin_fmaf(cp[t], P, d[t]);
            v4f* q = reinterpret_cast<v4f*>(
                yrow + tile * TILE + 16 * col + 8 * half);
            v4f lo = {o[0], o[1], o[2], o[3]};
            v4f hi = {o[4], o[5], o[6], o[7]};
            __builtin_nontemporal_store(lo, q);
            __builtin_nontemporal_store(hi, q + 1);
        }

        // ---- Rotate the software pipeline.
#pragma unroll
        for (int r = 0; r < 4; ++r) bcur[r] = bnxt[r];
    }
}

extern "C" void kernel_launch(void* const* d_in, const int* in_sizes, int n_in,
                              void* d_out, int out_size, void* d_ws, size_t ws_size,
                              hipStream_t stream) {
    (void)in_sizes; (void)n_in; (void)out_size; (void)d_ws; (void)ws_size;
    const float* x  = (const float*)d_in[0];
    float*       yo = (float*)d_out;
    deemph_scan_wmma<<<GRID_BLOCKS, BLOCK_THREADS, 0, stream>>>(x, yo);
}